// Net_5016521802331
// MI455X (gfx1250) — compile-verified
//
#include <hip/hip_runtime.h>
#include <cstdint>

typedef __attribute__((ext_vector_type(2))) float v2f;
typedef __attribute__((ext_vector_type(8))) float v8f;

#define KDIM        784
#define TSTEPS      100
#define NOUT        10
#define LDSW_STRIDE 786   // even (8B-aligned pairs), 786 % 64 = 18 -> conflict-free for 16 lanes

// ---------------------------------------------------------------------------
// Phase 1: proj[row, h] = x[row, :] . W[h, :] + bias[h]   (row = b*T + t)
// One wave per 16-row tile, K-loop of V_WMMA_F32_16X16X4_F32 (exact fp32 MAC).
// proj stored padded to stride 16 in d_ws.
// ---------------------------------------------------------------------------
__global__ __launch_bounds__(256) void snn_proj_wmma(
    const float* __restrict__ x, const float* __restrict__ W,
    const float* __restrict__ bias, float* __restrict__ proj, int num_rows)
{
    // Stage W^T operand in LDS, 16 rows (rows 10..15 zero padding).
    __shared__ float ldsW[16 * LDSW_STRIDE];
    for (int i = threadIdx.x; i < 16 * KDIM; i += 256) {
        int n = i / KDIM;
        int k = i - n * KDIM;
        ldsW[n * LDSW_STRIDE + k] = (n < NOUT) ? W[i] : 0.0f;
    }
    __syncthreads();

    const int lane = threadIdx.x & 31;
    const int wave = threadIdx.x >> 5;
    const int m    = lane & 15;   // A: row-in-tile, B/C/D: column N
    const int hi   = lane >> 4;   // selects K pair {0,1} vs {2,3}

    const int tile = blockIdx.x * 8 + wave;
    const int row0 = tile * 16;
    if (row0 >= num_rows) return;

    int rowA = row0 + m;
    if (rowA >= num_rows) rowA = num_rows - 1;   // clamp (harmless duplicate work)

    const float* xrow = x + (size_t)rowA * KDIM + 2 * hi;            // A fragment source
    const float* wrow = ldsW + (size_t)m * LDSW_STRIDE + 2 * hi;     // B fragment source

    v8f acc = {0.f, 0.f, 0.f, 0.f, 0.f, 0.f, 0.f, 0.f};

#pragma unroll 4
    for (int k0 = 0; k0 < KDIM; k0 += 4) {
        v2f a  = *(const v2f*)(xrow + k0);   // x[row][k0 + 2*hi + {0,1}]
        v2f bf = *(const v2f*)(wrow + k0);   // W[m][k0 + 2*hi + {0,1}]  (B = W^T)
        acc = __builtin_amdgcn_wmma_f32_16x16x4_f32(
            /*neg_a=*/false, a, /*neg_b=*/false, bf,
            /*c_mod=*/(short)0, acc, /*reuse_a=*/false, /*reuse_b=*/false);
    }

    const float bb = (m < NOUT) ? bias[m] : 0.0f;

    // C/D layout: VGPR r -> M = r + 8*hi, N = m. Store padded stride-16 rows.
    float* prow = proj + (size_t)row0 * 16 + m;
#pragma unroll
    for (int r = 0; r < 8; ++r) {
        int mm = r + 8 * hi;
        prow[(size_t)mm * 16] = acc[r] + bb;
    }
}

// ---------------------------------------------------------------------------
// Phase 2: sequential LIF scan per (b, h). Writes spk_rec/mem_rec (T+1, B, 10)
// including the zero t=0 slabs the reference prepends.
// new_mem uses the OLD syn; rst == out (both (mem-1)>0), as in the reference.
// ---------------------------------------------------------------------------
__global__ __launch_bounds__(256) void snn_scan(
    const float* __restrict__ proj, float* __restrict__ spk_rec,
    float* __restrict__ mem_rec, int B)
{
    const int tid = blockIdx.x * 256 + threadIdx.x;
    const int b = tid >> 4;
    const int h = tid & 15;
    if (b >= B || h >= NOUT) return;

    const float ALPHA = 0.81873075307798182f;  // exp(-0.001/0.005)
    const float BETA  = 0.90483741803595957f;  // exp(-0.001/0.010)

    const size_t bh   = (size_t)b * NOUT + h;
    const size_t slab = (size_t)B * NOUT;

    spk_rec[bh] = 0.0f;   // t = 0 slabs are zeros
    mem_rec[bh] = 0.0f;

    const float* p = proj + (size_t)b * TSTEPS * 16 + h;
    float syn = 0.0f, mem = 0.0f;

    for (int t = 0; t < TSTEPS; ++t) {
        float inp  = p[(size_t)t * 16];
        float mthr = mem - 1.0f;
        float out  = (mthr > 0.0f) ? 1.0f : 0.0f;
        float nsyn = ALPHA * syn + inp;
        float nmem = BETA * mem + syn - out;   // old syn, rst == out
        size_t o = (size_t)(t + 1) * slab + bh;
        spk_rec[o] = out;
        mem_rec[o] = nmem;
        syn = nsyn;
        mem = nmem;
    }
}

// ---------------------------------------------------------------------------
extern "C" void kernel_launch(void* const* d_in, const int* in_sizes, int n_in,
                              void* d_out, int out_size, void* d_ws, size_t ws_size,
                              hipStream_t stream)
{
    (void)n_in; (void)out_size; (void)ws_size;

    const float* x    = (const float*)d_in[0];   // (B, 100, 784)
    const float* W    = (const float*)d_in[1];   // (10, 784)
    const float* bias = (const float*)d_in[2];   // (10,)
    float*       out  = (float*)d_out;

    const int B        = in_sizes[0] / (TSTEPS * KDIM);
    const int num_rows = B * TSTEPS;
    const int tiles    = (num_rows + 15) / 16;

    float* proj = (float*)d_ws;                  // tiles*16*16 floats (~26 MB for B=4096)

    snn_proj_wmma<<<dim3((tiles + 7) / 8), 256, 0, stream>>>(x, W, bias, proj, num_rows);

    float* spk  = out;
    float* memr = out + (size_t)(TSTEPS + 1) * B * NOUT;
    const int nthr = B * 16;
    snn_scan<<<dim3((nthr + 255) / 256), 256, 0, stream>>>(proj, spk, memr, B);
}